// Chfloss_87823491268890
// MI455X (gfx1250) — compile-verified
//
#include <hip/hip_runtime.h>
#include <math.h>

// CHF loss on MI455X (gfx1250, wave32).
// Separable formulation: derived[b,u,v] = sum_{h,w} e^{i(fu*y_h + fv*x_w)} dnn[b,h,w]
//   Stage 1: T[b,h,v]   = sum_w dnn[b,h,w] * e^{i fv x_w}        (f32 WMMA 16x16x4)
//   Stage 2: D[b,u,v]   = sum_h e^{i fu y_h} * T[b,h,v]          (f32 WMMA 16x16x4, complex)
//   Stage 3: loss = sum_b sqrt(sum |D - chf|^2) * 0.05 / 16
// Templates generated analytically in registers with __sincosf (never materialized).

typedef __attribute__((ext_vector_type(2))) float v2f;
typedef __attribute__((ext_vector_type(8))) float v8f;

#define WMMA_F32X4(A, B, C) \
  __builtin_amdgcn_wmma_f32_16x16x4_f32(false, (A), false, (B), (short)0, (C), false, false)

// ---------------- Stage 1 ----------------
// Per batch b: out tiles over (h:128, v:64). 8x4 = 32 tiles/batch, 512 waves total.
// Each wave: A = dnn[b, h-tile, k..k+3] (16x4), B = cos/sin template (4x16), K loop over w.
__global__ __launch_bounds__(256) void chf_stage1(const float* __restrict__ dnn,
                                                  float* __restrict__ Tr,
                                                  float* __restrict__ Ti) {
  const int lane = threadIdx.x & 31;
  const int gw   = blockIdx.x * 8 + (threadIdx.x >> 5);   // 0..511
  const int b    = gw >> 5;
  const int rem  = gw & 31;
  const int ht   = rem >> 2;       // h tile 0..7
  const int vt   = rem & 3;        // v tile 0..3
  const int m    = lane & 15;
  const int koff = (lane >> 4) << 1;   // lanes 16-31 hold K+2,K+3
  const int h    = ht * 16 + m;        // A-matrix row (M)
  const int v    = vt * 16 + m;        // B-matrix col (N)
  const float fv = (float)(v - 32) * 0.05f;
  const float* arow = dnn + (b * 128 + h) * 128;

  v8f accr = {}; v8f acci = {};
  for (int k0 = 0; k0 < 128; k0 += 4) {
    const int w0 = k0 + koff;
    float s0, c0, s1, c1;
    __sincosf(fv * ((float)w0 + 0.5f), &s0, &c0);
    __sincosf(fv * ((float)w0 + 1.5f), &s1, &c1);
    v2f a;  a.x  = arow[w0]; a.y  = arow[w0 + 1];
    v2f bc; bc.x = c0;       bc.y = c1;
    v2f bs; bs.x = s0;       bs.y = s1;
    accr = WMMA_F32X4(a, bc, accr);
    acci = WMMA_F32X4(a, bs, acci);
  }

  const int hiM = (lane >> 4) << 3;   // C/D: lanes 16-31 hold M+8
  for (int r = 0; r < 8; ++r) {
    const int hr  = ht * 16 + r + hiM;
    const int idx = (b * 128 + hr) * 64 + v;
    Tr[idx] = accr[r];
    Ti[idx] = acci[r];
  }
}

// ---------------- Stage 2 ----------------
// Per batch b: out tiles over (u:64, v:64). 4x4 = 16 tiles/batch, 256 waves total.
// Complex multiply-accumulate: Re += cy*Tr - sy*Ti ; Im += sy*Tr + cy*Ti.
// (f32 WMMA has no A-negate, so -sy is built into the fragment.)
__global__ __launch_bounds__(256) void chf_stage2(const float* __restrict__ Tr,
                                                  const float* __restrict__ Ti,
                                                  const float* __restrict__ chf,
                                                  float* __restrict__ partial) {
  const int lane = threadIdx.x & 31;
  const int gw   = blockIdx.x * 8 + (threadIdx.x >> 5);   // 0..255
  const int b    = gw >> 4;
  const int rem  = gw & 15;
  const int ut   = rem >> 2;
  const int vt   = rem & 3;
  const int m    = lane & 15;
  const int koff = (lane >> 4) << 1;
  const int u    = ut * 16 + m;        // A row (M)
  const int v    = vt * 16 + m;        // B col (N)
  const float fu = (float)(u - 32) * 0.05f;
  const float* trb = Tr + b * 128 * 64;
  const float* tib = Ti + b * 128 * 64;

  v8f accre = {}; v8f accim = {};
  for (int h0 = 0; h0 < 128; h0 += 4) {
    const int hh = h0 + koff;
    float s0, c0, s1, c1;
    __sincosf(fu * ((float)hh + 0.5f), &s0, &c0);
    __sincosf(fu * ((float)hh + 1.5f), &s1, &c1);
    v2f ac;  ac.x  = c0;  ac.y  = c1;
    v2f as;  as.x  = s0;  as.y  = s1;
    v2f ans; ans.x = -s0; ans.y = -s1;
    v2f br;  br.x = trb[hh * 64 + v]; br.y = trb[(hh + 1) * 64 + v];
    v2f bi;  bi.x = tib[hh * 64 + v]; bi.y = tib[(hh + 1) * 64 + v];
    accre = WMMA_F32X4(ac,  br, accre);
    accre = WMMA_F32X4(ans, bi, accre);
    accim = WMMA_F32X4(as,  br, accim);
    accim = WMMA_F32X4(ac,  bi, accim);
  }

  const int hiM = (lane >> 4) << 3;
  float local = 0.f;
  for (int r = 0; r < 8; ++r) {
    const int ur   = ut * 16 + r + hiM;
    const int cidx = ((b * 64 + ur) * 64 + v) * 2;     // chf layout (B,64,64,2)
    const float dre = accre[r] - chf[cidx];
    const float dim = accim[r] - chf[cidx + 1];
    local += dre * dre + dim * dim;
  }
  // wave32 reduction; deterministic per-wave slot (no float atomics)
  for (int off = 16; off > 0; off >>= 1) local += __shfl_down(local, off, 32);
  if (lane == 0) partial[b * 16 + rem] = local;
}

// ---------------- Stage 3 ----------------
__global__ void chf_finalize(const float* __restrict__ partial, float* __restrict__ out) {
  const int lane = threadIdx.x & 31;
  float s = 0.f;
  if (lane < 16) {
    float acc = 0.f;
    for (int i = 0; i < 16; ++i) acc += partial[lane * 16 + i];
    s = sqrtf(acc);
  }
  for (int off = 16; off > 0; off >>= 1) s += __shfl_down(s, off, 32);
  if (lane == 0) out[0] = s * (0.05f / 16.0f);   // * CHF_TIK / B
}

extern "C" void kernel_launch(void* const* d_in, const int* in_sizes, int n_in,
                              void* d_out, int out_size, void* d_ws, size_t ws_size,
                              hipStream_t stream) {
  (void)in_sizes; (void)n_in; (void)out_size; (void)ws_size;
  const float* dnn = (const float*)d_in[0];   // (16,128,128) f32
  const float* chf = (const float*)d_in[1];   // (16,64,64,2) f32

  float* ws      = (float*)d_ws;
  float* Tr      = ws;                         // 16*128*64 floats
  float* Ti      = ws + 16 * 128 * 64;         // 16*128*64 floats
  float* partial = ws + 2 * 16 * 128 * 64;     // 256 floats

  chf_stage1<<<64, 256, 0, stream>>>(dnn, Tr, Ti);           // 512 waves
  chf_stage2<<<32, 256, 0, stream>>>(Tr, Ti, chf, partial);  // 256 waves
  chf_finalize<<<1, 32, 0, stream>>>(partial, (float*)d_out);
}